// Model_6571299963070
// MI455X (gfx1250) — compile-verified
//
#include <hip/hip_runtime.h>
#include <hip/hip_bf16.h>

#define N_USERS  200000
#define N_MOVIES 80000
#define N_TOT    (N_USERS + N_MOVIES)   // 280000
#define H        64
#define F_MOVIE  20
#define NE       1250000
#define NEL      500000

typedef __attribute__((ext_vector_type(2))) float v2f;
typedef __attribute__((ext_vector_type(8))) float v8f;

// ---------------------------------------------------------------------------
// x = concat(user_emb, movie_x @ lin_W.T + lin_b + movie_emb)
// ---------------------------------------------------------------------------
__global__ void __launch_bounds__(256)
k_build_x(const float* __restrict__ user_emb,
          const float* __restrict__ movie_x,
          const float* __restrict__ linW,
          const float* __restrict__ linb,
          const float* __restrict__ movie_emb,
          float* __restrict__ x)
{
    long i = (long)blockIdx.x * blockDim.x + threadIdx.x;
    if (i >= (long)N_TOT * H) return;
    long n = i >> 6;
    int  h = (int)(i & 63);
    if (n < N_USERS) {
        x[i] = user_emb[i];
    } else {
        long m = n - N_USERS;
        const float* mx = movie_x + m * F_MOVIE;
        const float* w  = linW + (long)h * F_MOVIE;
        float s = linb[h] + movie_emb[m * H + h];
        #pragma unroll
        for (int k = 0; k < F_MOVIE; ++k) s += mx[k] * w[k];
        x[i] = s;
    }
}

__global__ void __launch_bounds__(256)
k_zero(float* __restrict__ p, long n)
{
    long i = (long)blockIdx.x * blockDim.x + threadIdx.x;
    if (i < n) p[i] = 0.0f;
}

// degree (edge-topology only; computed once, reused for both layers)
__global__ void __launch_bounds__(256)
k_deg(const int* __restrict__ dst, float* __restrict__ deg)
{
    int e = blockIdx.x * blockDim.x + threadIdx.x;
    if (e < NE) atomicAdd(&deg[dst[e]], 1.0f);
}

__global__ void __launch_bounds__(256)
k_recip(const float* __restrict__ deg, float* __restrict__ recip)
{
    int i = blockIdx.x * blockDim.x + threadIdx.x;
    if (i < N_TOT) recip[i] = 1.0f / fmaxf(deg[i], 1.0f);
}

// scatter-add: agg[dst] += x[src]
// One thread per (edge, 4-col group): coalesced b128 row loads, 4 f32 atomics.
// x and agg are L2-resident (72 MB each < 192 MB L2).
__global__ void __launch_bounds__(256)
k_scatter(const float4* __restrict__ x4,
          const int* __restrict__ src, const int* __restrict__ dst,
          float* __restrict__ agg)
{
    long i = (long)blockIdx.x * blockDim.x + threadIdx.x;
    if (i >= (long)NE * (H / 4)) return;
    long e = i >> 4;
    int  q = (int)(i & 15);
    long s = (long)src[e];
    long d = (long)dst[e];
    float4 v = x4[s * (H / 4) + q];
    float* a = agg + d * H + q * 4;
    atomicAdd(a + 0, v.x);
    atomicAdd(a + 1, v.y);
    atomicAdd(a + 2, v.z);
    atomicAdd(a + 3, v.w);
}

// ---------------------------------------------------------------------------
// out = act( (agg*recip) @ Wl.T + b + x @ Wr.T )
// One wave computes a 16-row x 64-col strip via V_WMMA_F32_16X16X4_F32.
// Weights staged in LDS (2 x 16 KB).
// ---------------------------------------------------------------------------
template <bool RELU>
__global__ void __launch_bounds__(128)
k_sage_gemm(const float* __restrict__ xin,
            const float* __restrict__ agg,
            const float* __restrict__ recip,
            const float* __restrict__ Wl,
            const float* __restrict__ bias,
            const float* __restrict__ Wr,
            float* __restrict__ xout)
{
    __shared__ float lWl[H * H];
    __shared__ float lWr[H * H];
    for (int i = threadIdx.x; i < H * H; i += 128) {
        lWl[i] = Wl[i];
        lWr[i] = Wr[i];
    }
    __syncthreads();

    const int wave = blockIdx.x * (blockDim.x >> 5) + (threadIdx.x >> 5);
    const int lane = threadIdx.x & 31;
    const int ln   = lane & 15;   // position within half-wave
    const int hi   = lane >> 4;   // 0: K/M low half, 1: high half
    const long m0  = (long)wave * 16;

    const float* aggRow = agg + (m0 + ln) * H;
    const float* xRow   = xin + (m0 + ln) * H;
    const float  rc     = recip[m0 + ln];

    v8f acc[4];
    #pragma unroll
    for (int ct = 0; ct < 4; ++ct) {
        float bv = bias[ct * 16 + ln];
        #pragma unroll
        for (int r = 0; r < 8; ++r) acc[ct][r] = bv;
    }

    const int klane = 2 * hi;     // this lane's K sub-offset within a K=4 block
    #pragma unroll
    for (int kb = 0; kb < H / 4; ++kb) {
        const int ko = kb * 4 + klane;
        v2f am = *(const v2f*)(aggRow + ko);
        am *= rc;                                   // fused mean = agg / deg
        v2f ax = *(const v2f*)(xRow + ko);
        #pragma unroll
        for (int ct = 0; ct < 4; ++ct) {
            v2f bl = *(const v2f*)(&lWl[(ct * 16 + ln) * H + ko]);
            v2f br = *(const v2f*)(&lWr[(ct * 16 + ln) * H + ko]);
            acc[ct] = __builtin_amdgcn_wmma_f32_16x16x4_f32(
                false, am, false, bl, (short)0, acc[ct], false, false);
            acc[ct] = __builtin_amdgcn_wmma_f32_16x16x4_f32(
                false, ax, false, br, (short)0, acc[ct], false, false);
        }
    }

    // C/D layout: VGPR r -> row m0 + r + 8*hi, col = ct*16 + ln
    #pragma unroll
    for (int ct = 0; ct < 4; ++ct) {
        #pragma unroll
        for (int r = 0; r < 8; ++r) {
            float v = acc[ct][r];
            if (RELU) v = fmaxf(v, 0.0f);
            xout[(m0 + r + 8 * hi) * H + ct * 16 + ln] = v;
        }
    }
}

// ---------------------------------------------------------------------------
// out[i] = dot(x[u_i], x[N_USERS + m_i])   one wave32 per label edge
// lane reads a float2 of each operand -> single global_load_b64 each
// ---------------------------------------------------------------------------
__global__ void __launch_bounds__(256)
k_dot(const float* __restrict__ x,
      const int* __restrict__ ui, const int* __restrict__ mi,
      float* __restrict__ out)
{
    int wave = blockIdx.x * (blockDim.x >> 5) + (threadIdx.x >> 5);
    int lane = threadIdx.x & 31;
    if (wave >= NEL) return;
    long u = (long)ui[wave];
    long m = (long)N_USERS + (long)mi[wave];
    const float2* ur = (const float2*)(x + u * H);
    const float2* mr = (const float2*)(x + m * H);
    float2 a = ur[lane];
    float2 b = mr[lane];
    float s = a.x * b.x + a.y * b.y;
    #pragma unroll
    for (int off = 16; off > 0; off >>= 1)
        s += __shfl_down(s, off, 32);
    if (lane == 0) out[wave] = s;
}

// ---------------------------------------------------------------------------
extern "C" void kernel_launch(void* const* d_in, const int* in_sizes, int n_in,
                              void* d_out, int out_size, void* d_ws, size_t ws_size,
                              hipStream_t stream)
{
    const float* movie_x   = (const float*)d_in[0];
    const float* user_emb  = (const float*)d_in[1];
    const float* movie_emb = (const float*)d_in[2];
    const float* lin_W     = (const float*)d_in[3];
    const float* lin_b     = (const float*)d_in[4];
    const float* W1l       = (const float*)d_in[5];
    const float* b1        = (const float*)d_in[6];
    const float* W1r       = (const float*)d_in[7];
    const float* W2l       = (const float*)d_in[8];
    const float* b2        = (const float*)d_in[9];
    const float* W2r       = (const float*)d_in[10];
    const int*   eidx      = (const int*)d_in[11];   // (2, E) flat
    const int*   elidx     = (const int*)d_in[12];   // (2, EL) flat
    float*       out       = (float*)d_out;

    const int* src = eidx;
    const int* dst = eidx + NE;
    const int* eu  = elidx;
    const int* em  = elidx + NEL;

    const size_t NX = (size_t)N_TOT * H;
    float* xA    = (float*)d_ws;          // x0, later reused for x2
    float* xB    = xA + NX;               // x1
    float* agg   = xB + NX;
    float* deg   = agg + NX;
    float* recip = deg + N_TOT;

    const long nx    = (long)NX;
    const long nscat = (long)NE * (H / 4);

    dim3 b256(256);
    dim3 gX((unsigned)((nx + 255) / 256));
    dim3 gScat((unsigned)((nscat + 255) / 256));
    dim3 gE((NE + 255) / 256);
    dim3 gN((N_TOT + 255) / 256);

    // build x0
    k_build_x<<<gX, b256, 0, stream>>>(user_emb, movie_x, lin_W, lin_b, movie_emb, xA);

    // degree + reciprocal (topology only; once)
    k_zero<<<gN, b256, 0, stream>>>(deg, N_TOT);
    k_deg<<<gE, b256, 0, stream>>>(dst, deg);
    k_recip<<<gN, b256, 0, stream>>>(deg, recip);

    const int waves  = N_TOT / 16;             // 17500 row-strips
    dim3 bGemm(128);                           // 4 waves / block
    dim3 gGemm(waves / 4);

    // layer 1: agg = scatter(x0); x1 = relu(mean@W1l.T + b1 + x0@W1r.T)
    k_zero<<<gX, b256, 0, stream>>>(agg, nx);
    k_scatter<<<gScat, b256, 0, stream>>>((const float4*)xA, src, dst, agg);
    k_sage_gemm<true><<<gGemm, bGemm, 0, stream>>>(xA, agg, recip, W1l, b1, W1r, xB);

    // layer 2: agg = scatter(x1); x2 = mean@W2l.T + b2 + x1@W2r.T  (into xA)
    k_zero<<<gX, b256, 0, stream>>>(agg, nx);
    k_scatter<<<gScat, b256, 0, stream>>>((const float4*)xB, src, dst, agg);
    k_sage_gemm<false><<<gGemm, bGemm, 0, stream>>>(xB, agg, recip, W2l, b2, W2r, xA);

    // link prediction dot products
    dim3 gDot((NEL * 32 + 255) / 256);
    k_dot<<<b256.x ? gDot : gDot, b256, 0, stream>>>(xA, eu, em, out);
}